// LocallyConnected2D_69793218560044
// MI455X (gfx1250) — compile-verified
//
#include <hip/hip_runtime.h>
#include <cstdint>

// ---- ext-vector types for WMMA / TDM builtins ----
typedef __attribute__((ext_vector_type(2))) float        v2f;
typedef __attribute__((ext_vector_type(8))) float        v8f;
typedef __attribute__((ext_vector_type(4))) unsigned int v4u;
typedef __attribute__((ext_vector_type(8))) int          v8i;
typedef __attribute__((ext_vector_type(4))) int          v4i;

// ---- problem geometry ----
constexpr int NB   = 64;
constexpr int HIN  = 512, WIN = 512;
constexpr int HOUT = 504, WOUT = 504;
constexpr int SH = 12, SW = 12;        // output tile per block
constexpr int XT = 20;                 // SH+8 (x tile edge incl. halo)
constexpr int XSN = XT * XT;           // 400 floats per batch image tile
constexpr int KWIN = 144;              // 12*12 K window for a 4x4 subtile
constexpr int NPIX = SH * SW;          // 144 pixels per block

// ---------------------------------------------------------------------------
// Tensor Data Mover: 3D tile load Global -> LDS (data_size = 4 bytes).
//   X (innermost, contiguous) = td/tl 0, Y stride st0, Z stride st1.
//   LDS receives the tile fully packed: [Z][Y][X].
// D# bit layout per CDNA5 ISA ch. 8 (group0 128b, group1 256b, group2/3 128b).
// ---------------------------------------------------------------------------
__device__ __forceinline__ void tdm_load_3d(uint32_t lds_byte_addr, uint64_t gaddr,
                                            uint32_t td0, uint32_t td1, uint32_t td2,
                                            uint64_t st0, uint64_t st1,
                                            uint32_t tl0, uint32_t tl1, uint32_t tl2)
{
    v4u g0;
    g0.x = 1u;                                   // count=1, user mode, no gather
    g0.y = lds_byte_addr;                        // lds_addr (bytes)
    g0.z = (uint32_t)gaddr;                      // global_addr[31:0]
    g0.w = (uint32_t)(gaddr >> 32) | (2u << 30); // global_addr[56:32] | type=2

    v8i g1;
    g1[0] = (int)(2u << 16);                                            // data_size=4B
    g1[1] = (int)((td0 & 0xffffu) << 16);                               // tensor_dim0 lo
    g1[2] = (int)((td0 >> 16) | ((td1 & 0xffffu) << 16));               // dim0 hi | dim1 lo
    g1[3] = (int)((td1 >> 16) | (tl0 << 16));                           // dim1 hi | tile_dim0
    g1[4] = (int)(tl1 | (tl2 << 16));                                   // tile_dim1 | tile_dim2
    g1[5] = (int)(uint32_t)st0;                                         // dim0_stride lo
    g1[6] = (int)((uint32_t)(st0 >> 32) | ((uint32_t)(st1 & 0xffffu) << 16));
    g1[7] = (int)(uint32_t)(st1 >> 16);                                 // dim1_stride hi

    v4i g2 = { (int)td2, 0, 0, 0 };              // tensor_dim2; no dim3
    v4i g3 = { 0, 0, 0, 0 };

#if __clang_major__ >= 23
    v8i g4 = { 0, 0, 0, 0, 0, 0, 0, 0 };
    __builtin_amdgcn_tensor_load_to_lds(g0, g1, g2, g3, g4, 0);
#else
    __builtin_amdgcn_tensor_load_to_lds(g0, g1, g2, g3, 0);
#endif
}

// ---------------------------------------------------------------------------
// Locally-connected 2D as banded GEMM on v_wmma_f32_16x16x4_f32.
//  grid = (42, 42); block = 128 threads (4 waves, wave w = batch [16w,16w+16)).
//  Block tile: 12x12 output pixels x 64 batch; 9 sub-tiles of 4x4 pixels.
//  Per sub-tile: D[16n x 16pix] = sum over K=144 (12x12 patch window) of
//    A[n, k] = x[n, h0+sdh + k/12, w0+sdw + k%12]    (LDS xs, TDM-staged)
//    B[k, p] = W[pix p, k/12 - p/4, k%12 - p%4] or 0 (LDS wpad, zero-padded)
// ---------------------------------------------------------------------------
__global__ __launch_bounds__(128)
void lc2d_wmma_kernel(const float* __restrict__ x,
                      const float* __restrict__ w,
                      const float* __restrict__ bias,
                      float* __restrict__ out)
{
    __shared__ alignas(16) float xs[NB * XSN];      // 102400 B  [n][20][20]
    __shared__ alignas(16) float ws[NPIX * 81];     //  46656 B  [dh][dw][81]
    __shared__ alignas(16) float wpad[16 * KWIN];   //   9216 B  [pix][k] zero-padded

    const int h0   = blockIdx.y * SH;
    const int w0   = blockIdx.x * SW;
    const int tid  = threadIdx.x;
    const int lane = tid & 31;
    const int wv   = tid >> 5;        // wave id = batch m-tile
    const int col  = lane & 15;       // WMMA N column (pixel) / A row (batch)
    const int h16  = lane >> 4;       // lane half: selects K pair {0,1} vs {2,3}

    // ---- stage x tile and weight tile with the Tensor Data Mover (wave 0) ----
    if (tid < 32) {
        const uint32_t xs_lds = (uint32_t)(uintptr_t)&xs[0];
        const uint32_t ws_lds = (uint32_t)(uintptr_t)&ws[0];
        const uint64_t xg = (uint64_t)(uintptr_t)x + (uint64_t)(h0 * WIN + w0) * 4u;
        const uint64_t wg = (uint64_t)(uintptr_t)w + (uint64_t)(h0 * WOUT + w0) * 81u * 4u;
        // x: tile [c=20][r=20][n=64], strides row=512, image=512*512
        tdm_load_3d(xs_lds, xg, WIN, HIN, NB,
                    (uint64_t)WIN, (uint64_t)HIN * WIN,
                    XT, XT, NB);
        // weight: one output row = 12 pixels * 81 taps = 972 contiguous floats;
        // tile [972][12 rows][1], row stride = 504*81
        tdm_load_3d(ws_lds, wg, (uint32_t)(WOUT * 81), HOUT, 1,
                    (uint64_t)WOUT * 81, 0ull,
                    (uint32_t)(SW * 81), SH, 1);
        __builtin_amdgcn_s_wait_tensorcnt(0);
    }
    __syncthreads();

    for (int st = 0; st < 9; ++st) {
        const int sdh = (st / 3) * 4;
        const int sdw = (st % 3) * 4;

        __syncthreads();   // previous sub-tile's wpad readers done
        // ---- build zero-padded B panel for this 4x4 pixel sub-tile ----
        for (int e = tid; e < 16 * KWIN; e += 128) {   // 2304 = 18 * 128, exact
            const int p = e / KWIN;          // pixel 0..15  (dh_l = p>>2, dw_l = p&3)
            const int k = e - p * KWIN;      // 0..143
            const int r = k / 12;
            const int c = k - r * 12;
            const int i = r - (p >> 2);
            const int j = c - (p & 3);
            float v = 0.0f;
            if ((unsigned)i < 9u && (unsigned)j < 9u)
                v = ws[((sdh + (p >> 2)) * SW + (sdw + (p & 3))) * 81 + i * 9 + j];
            wpad[e] = v;
        }
        __syncthreads();

        // ---- K loop: 36 steps of v_wmma_f32_16x16x4_f32, 2 accumulators ----
        v8f acc0 = {0.f,0.f,0.f,0.f,0.f,0.f,0.f,0.f};
        v8f acc1 = {0.f,0.f,0.f,0.f,0.f,0.f,0.f,0.f};
        int c2   = 2 * h16;                                    // k0 % 12 tracker
        int aidx = (wv * 16 + col) * XSN + sdh * XT + sdw + c2; // float index into xs
        const int bbase = col * KWIN + 2 * h16;                 // float index into wpad
#pragma unroll
        for (int t = 0; t < 36; ++t) {
            v2f av = *reinterpret_cast<const v2f*>(xs + aidx);           // ds_load_b64
            v2f bv = *reinterpret_cast<const v2f*>(wpad + bbase + 4 * t); // ds_load_b64
            if (t & 1)
                acc1 = __builtin_amdgcn_wmma_f32_16x16x4_f32(
                           false, av, false, bv, (short)0, acc1, false, false);
            else
                acc0 = __builtin_amdgcn_wmma_f32_16x16x4_f32(
                           false, av, false, bv, (short)0, acc0, false, false);
            // advance k0 by 4 within the 12-wide patch window (branch-free)
            c2 += 4;
            aidx += 4;
            const int wrap = (c2 >= 12) ? 1 : 0;
            c2   -= wrap * 12;
            aidx += wrap * 8;          // row stride 20, window width 12
        }
        const v8f acc = acc0 + acc1;

        // ---- bias + ReLU + store (D: vreg r -> M = r + 8*h16, N = col) ----
        const int   hp = h0 + sdh + (col >> 2);
        const int   wp = w0 + sdw + (col & 3);
        const float bb = bias[hp * WOUT + wp];
        const size_t obase = (size_t)hp * WOUT + wp;
#pragma unroll
        for (int r8 = 0; r8 < 8; ++r8) {
            const int n = wv * 16 + h16 * 8 + r8;
            float v = acc[r8] + bb;
            v = v > 0.f ? v : 0.f;
            __builtin_nontemporal_store(v, out + (size_t)n * (HOUT * WOUT) + obase);
        }
    }
}

extern "C" void kernel_launch(void* const* d_in, const int* in_sizes, int n_in,
                              void* d_out, int out_size, void* d_ws, size_t ws_size,
                              hipStream_t stream) {
    (void)in_sizes; (void)n_in; (void)out_size; (void)d_ws; (void)ws_size;
    const float* x    = (const float*)d_in[0];
    const float* wgt  = (const float*)d_in[1];
    const float* bias = (const float*)d_in[2];
    float* out = (float*)d_out;
    dim3 grid(WOUT / SW, HOUT / SH);   // 42 x 42 blocks, exact tiling
    lc2d_wmma_kernel<<<grid, 128, 0, stream>>>(x, wgt, bias, out);
}